// ObjectLoss_58248346469109
// MI455X (gfx1250) — compile-verified
//
#include <hip/hip_runtime.h>

#define NUM_PIDS 20000
#define SCALE_F  100.0f

typedef float v2f __attribute__((ext_vector_type(2)));
typedef float v8f __attribute__((ext_vector_type(8)));

// ---------------------------------------------------------------- zero bins
__global__ void ol_zero_bins(float* __restrict__ ws, int n) {
  int i = blockIdx.x * blockDim.x + threadIdx.x;
  if (i < n) ws[i] = 0.0f;
}

// ------------------------------------------------- WMMA streaming accumulate
// One wave handles a tile of 16 hits. Squared diffs are laid out as the
// 16x4 f32 A matrix of V_WMMA_F32_16X16X4_F32 (lanes 0-15: K=0,1 in the two
// A VGPRs; lanes 16-31: K=2,3). B = all-ones so D[m][n] = sum_k A[m][k],
// i.e. the per-hit MSE, computed in full f32 by the matrix pipe.
__global__ __launch_bounds__(256) void ol_accum_wmma(
    const float* __restrict__ pred, const float* __restrict__ tp,
    const int*   __restrict__ pid,  const int*   __restrict__ rec,
    float* __restrict__ sums, float* __restrict__ cnts, int ntiles)
{
  const int lane   = threadIdx.x & 31;
  const int wave   = (int)((blockIdx.x * (long)blockDim.x + threadIdx.x) >> 5);
  const int nwaves = (int)(((long)gridDim.x * blockDim.x) >> 5);
  const int m   = lane & 15;   // hit row within tile (A-matrix M)
  const int hi  = lane >> 4;   // 0: K pair {0,1}/{4,5}, 1: K pair {2,3}/{6,7}
  const int sub = lane & 7;    // C-matrix VGPR index for extraction

  v2f ones; ones[0] = 1.0f; ones[1] = 1.0f;

  for (int t = wave; t < ntiles; t += nwaves) {
    const long base = (long)t << 4;
    const long hit  = base + m;
    const float* pp = pred + hit * 8 + hi * 2;
    const float* tt = tp   + hit * 8 + hi * 2;
    // coalesced b64 loads: lanes jointly cover 512B contiguous per tile in each of pred and tp
    float2 p0 = *(const float2*)(pp);
    float2 t0 = *(const float2*)(tt);
    float2 p1 = *(const float2*)(pp + 4);
    float2 t1 = *(const float2*)(tt + 4);
    float d0 = p0.x - t0.x, d1 = p0.y - t0.y;
    float d2 = p1.x - t1.x, d3 = p1.y - t1.y;
    v2f a0; a0[0] = d0 * d0; a0[1] = d1 * d1;   // params {0,1} / {2,3}
    v2f a1; a1[0] = d2 * d2; a1[1] = d3 * d3;   // params {4,5} / {6,7}

    float mse;
    bool  active;
    long  h2;
#if defined(__gfx1250__) && __has_builtin(__builtin_amdgcn_wmma_f32_16x16x4_f32)
    v8f c = {};
    c = __builtin_amdgcn_wmma_f32_16x16x4_f32(false, a0, false, ones,
                                              (short)0, c, false, false);
    c = __builtin_amdgcn_wmma_f32_16x16x4_f32(false, a1, false, ones,
                                              (short)0, c, false, false);
    // C/D layout: lanes 0-15 VGPR v = row M=v; lanes 16-31 VGPR v = row M=8+v.
    mse = c[0];
    mse = (sub == 1) ? c[1] : mse;
    mse = (sub == 2) ? c[2] : mse;
    mse = (sub == 3) ? c[3] : mse;
    mse = (sub == 4) ? c[4] : mse;
    mse = (sub == 5) ? c[5] : mse;
    mse = (sub == 6) ? c[6] : mse;
    mse = (sub == 7) ? c[7] : mse;
    active = (m < 8);                 // lanes 0-7 -> hits 0-7, 16-23 -> 8-15
    h2     = base + (hi << 3) + sub;
#else
    // Fallback: cross-half shuffle reduction (should not be needed on gfx1250)
    float partial = a0[0] + a0[1] + a1[0] + a1[1];
    mse    = partial + __shfl_xor(partial, 16, 32);
    active = (lane < 16);
    h2     = hit;
#endif
    if (active) {
      int p = pid[h2];
      if (rec[h2] > 0 && p > 0 && p < NUM_PIDS) {
        atomicAdd(&sums[p], mse);
        atomicAdd(&cnts[p], 1.0f);
      }
    }
  }
}

// ----------------------------------------------------------- scalar tail
__global__ void ol_tail(const float* __restrict__ pred, const float* __restrict__ tp,
                        const int* __restrict__ pid, const int* __restrict__ rec,
                        float* __restrict__ sums, float* __restrict__ cnts,
                        int start, int n)
{
  int i = start + blockIdx.x * blockDim.x + threadIdx.x;
  if (i < n) {
    float s = 0.0f;
    const float* pp = pred + (long)i * 8;
    const float* tt = tp + (long)i * 8;
#pragma unroll
    for (int d = 0; d < 8; ++d) { float df = pp[d] - tt[d]; s += df * df; }
    int p = pid[i];
    if (rec[i] > 0 && p > 0 && p < NUM_PIDS) {
      atomicAdd(&sums[p], s);
      atomicAdd(&cnts[p], 1.0f);
    }
  }
}

// ----------------------------------------------------------- finalize
__global__ __launch_bounds__(1024) void ol_finalize(
    const float* __restrict__ sums, const float* __restrict__ cnts,
    float* __restrict__ out)
{
  __shared__ float ssum[1024];
  __shared__ float skk[1024];
  float s = 0.0f, k = 0.0f;
  for (int i = threadIdx.x; i < NUM_PIDS; i += 1024) {
    float c = cnts[i];
    if (i > 0 && c > 0.0f) { s += sums[i] / c; k += 1.0f; }
  }
  ssum[threadIdx.x] = s;
  skk[threadIdx.x]  = k;
  __syncthreads();
  for (int off = 512; off > 0; off >>= 1) {
    if ((int)threadIdx.x < off) {
      ssum[threadIdx.x] += ssum[threadIdx.x + off];
      skk[threadIdx.x]  += skk[threadIdx.x + off];
    }
    __syncthreads();
  }
  if (threadIdx.x == 0)
    out[0] = (skk[0] > 0.0f) ? (SCALE_F * ssum[0] / skk[0]) : 0.0f;
}

// ----------------------------------------------------------- launcher
extern "C" void kernel_launch(void* const* d_in, const int* in_sizes, int n_in,
                              void* d_out, int out_size, void* d_ws, size_t ws_size,
                              hipStream_t stream) {
  // setup_inputs order: W, beta, H, pred, Y, particle_id, track_params, reconstructable
  const float* pred = (const float*)d_in[3];
  const int*   pid  = (const int*)d_in[5];
  const float* tp   = (const float*)d_in[6];
  const int*   rec  = (const int*)d_in[7];
  const int n = in_sizes[0];

  float* sums = (float*)d_ws;
  float* cnts = sums + NUM_PIDS;

  ol_zero_bins<<<(2 * NUM_PIDS + 255) / 256, 256, 0, stream>>>(sums, 2 * NUM_PIDS);

  const int ntiles = n >> 4;
  if (ntiles > 0) {
    const int waves_per_block = 256 / 32;
    long blocks = (ntiles + waves_per_block - 1) / waves_per_block;
    if (blocks > 32768) blocks = 32768;      // grid-stride covers the rest
    ol_accum_wmma<<<(int)blocks, 256, 0, stream>>>(pred, tp, pid, rec,
                                                   sums, cnts, ntiles);
  }
  const int tail0 = ntiles << 4;
  if (tail0 < n) {
    int rem = n - tail0;
    ol_tail<<<(rem + 255) / 256, 256, 0, stream>>>(pred, tp, pid, rec,
                                                   sums, cnts, tail0, n);
  }

  ol_finalize<<<1, 1024, 0, stream>>>(sums, cnts, (float*)d_out);
}